// MPSPeriodic_51771535786064
// MI455X (gfx1250) — compile-verified
//
#include <hip/hip_runtime.h>

typedef __attribute__((ext_vector_type(2))) float v2f;
typedef __attribute__((ext_vector_type(8))) float v8f;

#define BATCH 1024
#define LCH   64
#define PCH   2
#define DDIM  64
#define LSTR  66   // padded LDS row stride (even -> 8B alignment kept, breaks bank conflicts)

__global__ __launch_bounds__(128)
void mps_chain_kernel(const int* __restrict__ x,
                      const float* __restrict__ kern,
                      float* __restrict__ out)
{
    __shared__ float carry[2][DDIM * LSTR];
    __shared__ float bmatT[DDIM * LSTR];   // bmatT[n*LSTR + k] = B[k][n] (+I)
    __shared__ int   xrow[LCH];
    __shared__ float red[DDIM];

    const int b    = blockIdx.x;
    const int tid  = threadIdx.x;      // 0..127
    const int wave = tid >> 5;         // 0..3  -> output tile row band
    const int lane = tid & 31;

    // preload this batch's index row
    if (tid < LCH) xrow[tid] = x[b * LCH + tid];

    // carry[0] = Identity
    for (int idx = tid; idx < DDIM * DDIM; idx += 128) {
        int r = idx >> 6, c = idx & 63;
        carry[0][r * LSTR + c] = (r == c) ? 1.0f : 0.0f;
    }

    const int mBase   = wave * 16;         // band of output rows this wave owns
    const int mRow    = lane & 15;         // A-frag: M = lane%16
    const int kHi     = (lane >> 4) * 2;   // lanes 16..31 hold K+2 / K+3
    const int nCol    = lane & 15;         // B-frag: N = lane%16
    const int dRowOff = (lane >> 4) * 8;   // C/D: vgpr r -> row r (lo lanes) / r+8 (hi lanes)
    const int dCol    = lane & 15;

    int cur = 0;
    for (int l = 0; l < LCH; ++l) {
        __syncthreads();  // prior compute finished reading bmatT / carry buffers

        // stage selected matrix, transposed, with identity folded in
        const int sel = xrow[l];
        const float* Bg = kern + (size_t)(l * PCH + sel) * DDIM * DDIM;
        for (int idx = tid; idx < DDIM * DDIM; idx += 128) {
            int kk = idx >> 6, nn = idx & 63;
            float v = Bg[idx] + ((kk == nn) ? 1.0f : 0.0f);
            bmatT[nn * LSTR + kk] = v;
        }
        __syncthreads();

        const float* cbuf = carry[cur];
        v8f acc0 = {}, acc1 = {}, acc2 = {}, acc3 = {};

        #pragma unroll
        for (int kt = 0; kt < 16; ++kt) {
            const int k = kt * 4 + kHi;
            // A fragment: rows mBase..mBase+15 of carry, K chunk k..k+1 (contiguous)
            v2f a = *(const v2f*)(cbuf + (mBase + mRow) * LSTR + k);
            // B fragments: columns tj*16..tj*16+15, K chunk (contiguous in bmatT)
            v2f b0 = *(const v2f*)(bmatT + ( 0 + nCol) * LSTR + k);
            v2f b1 = *(const v2f*)(bmatT + (16 + nCol) * LSTR + k);
            v2f b2 = *(const v2f*)(bmatT + (32 + nCol) * LSTR + k);
            v2f b3 = *(const v2f*)(bmatT + (48 + nCol) * LSTR + k);
            acc0 = __builtin_amdgcn_wmma_f32_16x16x4_f32(false, a, false, b0,
                                                         (short)0, acc0, false, false);
            acc1 = __builtin_amdgcn_wmma_f32_16x16x4_f32(false, a, false, b1,
                                                         (short)0, acc1, false, false);
            acc2 = __builtin_amdgcn_wmma_f32_16x16x4_f32(false, a, false, b2,
                                                         (short)0, acc2, false, false);
            acc3 = __builtin_amdgcn_wmma_f32_16x16x4_f32(false, a, false, b3,
                                                         (short)0, acc3, false, false);
        }

        // write new carry into the other buffer (no race with readers of carry[cur])
        float* nbuf = carry[cur ^ 1];
        #pragma unroll
        for (int r = 0; r < 8; ++r) {
            const int row = (mBase + dRowOff + r) * LSTR;
            nbuf[row +  0 + dCol] = acc0[r];
            nbuf[row + 16 + dCol] = acc1[r];
            nbuf[row + 32 + dCol] = acc2[r];
            nbuf[row + 48 + dCol] = acc3[r];
        }
        cur ^= 1;
    }

    __syncthreads();
    if (tid < DDIM) red[tid] = carry[cur][tid * LSTR + tid];
    __syncthreads();
    if (tid == 0) {
        float s = 0.0f;
        #pragma unroll
        for (int i = 0; i < DDIM; ++i) s += red[i];
        out[b] = s;
    }
}

extern "C" void kernel_launch(void* const* d_in, const int* in_sizes, int n_in,
                              void* d_out, int out_size, void* d_ws, size_t ws_size,
                              hipStream_t stream) {
    (void)in_sizes; (void)n_in; (void)out_size; (void)d_ws; (void)ws_size;
    const int*   x    = (const int*)d_in[0];     // (B, L) int32
    const float* kern = (const float*)d_in[1];   // (L, P, D, D) float32
    float*       out  = (float*)d_out;           // (B,) float32
    mps_chain_kernel<<<BATCH, 128, 0, stream>>>(x, kern, out);
}